// DisjointDense_26113401159978
// MI455X (gfx1250) — compile-verified
//
#include <hip/hip_runtime.h>

// Problem constants (match the reference)
#define B_ROWS 4096
#define D_IN   256
#define D_OUT  256
#define NEXP   64

// LDS strides chosen for bank-conflict-free WMMA operand reads (64 banks):
// XS % 64 == 4 -> A read bank = (4*m + k) % 64, distinct across m=0..15
// WS % 64 == 4 -> B read banks consecutive across lanes
#define XS 260   // x tile row stride in floats (16 rows x 256 cols)
#define WS 68    // W panel row stride in floats (256 rows x 64 cols), set by TDM pad

typedef __attribute__((ext_vector_type(2))) float    v2f;
typedef __attribute__((ext_vector_type(8))) float    v8f;
typedef __attribute__((ext_vector_type(4))) unsigned v4u;
typedef __attribute__((ext_vector_type(8))) int      v8i;

__global__ void dd_zero_counts(int* __restrict__ counts) {
    counts[threadIdx.x] = 0;
}

// Decode one-hot selector, bucket rows per expert.
__global__ void dd_scatter(const float* __restrict__ onehot,
                           int* __restrict__ counts,
                           int* __restrict__ rowlist) {
    int b = blockIdx.x * blockDim.x + threadIdx.x;
    if (b >= B_ROWS) return;
    const float* row = onehot + (size_t)b * NEXP;
    int sel = 0;
    #pragma unroll
    for (int d = 0; d < NEXP; ++d) {
        if (row[d] > 0.5f) sel = d;
    }
    int pos = atomicAdd(&counts[sel], 1);
    rowlist[(size_t)sel * B_ROWS + pos] = b;
}

// Grouped GEMM: block = (expert e, 64-col group cg); 128 threads = 4 wave32.
// Wave 0 DMAs the full W[e][:, n_base:n_base+64] panel into LDS once via the
// Tensor Data Mover (with HW padding to a 68-float row stride), then the block
// loops over this expert's 16-row tiles: gather x rows -> LDS, 64 x
// v_wmma_f32_16x16x4_f32 over K=256, biased store.
__global__ __launch_bounds__(128) void dd_grouped_gemm(
        const float* __restrict__ x,
        const float* __restrict__ W,
        const float* __restrict__ Bw,
        const int*   __restrict__ counts,
        const int*   __restrict__ rowlist,
        float*       __restrict__ out) {
    __shared__ float wc[D_IN * WS];   // 256 x 68 floats = 69632 B
    __shared__ float xt[16 * XS];     // 16 x 260 floats  = 16640 B
    __shared__ int   srow[16];

    const int e    = blockIdx.x;       // expert
    const int cg   = blockIdx.y;       // column group (64 cols)
    const int tid  = threadIdx.x;      // 0..127
    const int lane = tid & 31;
    const int wave = tid >> 5;         // 0..3
    const int half = lane >> 4;        // lane group (0: lanes 0-15, 1: 16-31)
    const int l15  = lane & 15;

    const int n_base = cg * 64;
    const int n0     = n_base + wave * 16;       // this wave's output columns
    const int count  = counts[e];
    const int ntiles = (count + 15) >> 4;

    const float* We   = W + (size_t)e * D_IN * D_OUT;
    const float  bias = Bw[(size_t)e * D_OUT + n0 + l15];
    const int*   rl   = rowlist + (size_t)e * B_ROWS;

    // ---- TDM: one DMA of the 256x64 W panel into LDS (wave 0 issues) ----
    if (tid < 32) {
        unsigned long long ga = (unsigned long long)(const void*)(We + n_base);
        unsigned ga_lo = (unsigned)__builtin_amdgcn_readfirstlane((int)(unsigned)ga);
        unsigned ga_hi = (unsigned)__builtin_amdgcn_readfirstlane((int)(unsigned)(ga >> 32));
        unsigned lds   = (unsigned)(unsigned long long)(size_t)(void*)&wc[0];
        lds = (unsigned)__builtin_amdgcn_readfirstlane((int)lds);

        // D# group 0: count=1 | lds_addr | global_addr[56:0] | type=2
        v4u g0;
        g0[0] = 1u;
        g0[1] = lds;
        g0[2] = ga_lo;
        g0[3] = (ga_hi & 0x01FFFFFFu) | 0x80000000u;

        // D# group 1:
        //  d0: data_size=2 (4B) | pad_enable | pad_interval=5 (64 DW) | pad_amount=3 (4 DW)
        //  tensor_dim0=256, tensor_dim1=256, tile_dim0=64 (cols), tile_dim1=256 (rows)
        //  tensor_dim0_stride=256 elements
        v8i g1;
        g1[0] = (2 << 16) | (1 << 20) | (5 << 22) | (3 << 25);
        g1[1] = (int)(256u << 16);   // tensor_dim0[15:0] in bits[31:16]
        g1[2] = (int)(256u << 16);   // tensor_dim0 hi=0 | tensor_dim1[15:0]
        g1[3] = (int)(64u  << 16);   // tensor_dim1 hi=0 | tile_dim0=64
        g1[4] = 256;                 // tile_dim1=256 | tile_dim2=0
        g1[5] = 256;                 // tensor_dim0_stride lo32
        g1[6] = 0;
        g1[7] = 0;

        asm volatile("tensor_load_to_lds %0, %1" :: "s"(g0), "s"(g1) : "memory");
        __builtin_amdgcn_s_wait_tensorcnt(0);
    }
    __syncthreads();   // W panel resident in LDS for all waves

    for (int t = 0; t < ntiles; ++t) {
        __syncthreads();  // previous iteration's srow/xt readers are done
        if (tid < 16) {
            int idx = t * 16 + tid;
            srow[tid] = (idx < count) ? rl[idx] : -1;
        }
        __syncthreads();

        // Gather 16 x rows -> LDS. Issue all loads, then all stores.
        float4 tmp[8];
        #pragma unroll
        for (int j = 0; j < 8; ++j) {
            int idx = tid + j * 128;   // float4 index in [0,1024)
            int r   = idx >> 6;        // 64 float4 per row
            int c4  = idx & 63;
            int row = srow[r];
            tmp[j] = make_float4(0.f, 0.f, 0.f, 0.f);
            if (row >= 0) {
                tmp[j] = *(const float4*)(x + (size_t)row * D_IN + c4 * 4);
            }
        }
        #pragma unroll
        for (int j = 0; j < 8; ++j) {
            int idx = tid + j * 128;
            int r   = idx >> 6;
            int c4  = idx & 63;
            *(float4*)(&xt[r * XS + c4 * 4]) = tmp[j];
        }
        __syncthreads();

        v8f acc;
        #pragma unroll
        for (int i = 0; i < 8; ++i) acc[i] = 0.0f;

        // 64 x v_wmma_f32_16x16x4_f32 over K=256.
        // A (16x4): lane -> M = l15, K = half*2 + vgpr
        // B (4x16): lane -> N = l15, K = half*2 + vgpr
        #pragma unroll
        for (int k = 0; k < D_IN; k += 4) {
            int kk = k + half * 2;
            v2f a = *(const v2f*)(&xt[l15 * XS + kk]);   // 8B LDS read
            v2f b;
            b.x = wc[(kk + 0) * WS + wave * 16 + l15];
            b.y = wc[(kk + 1) * WS + wave * 16 + l15];
            acc = __builtin_amdgcn_wmma_f32_16x16x4_f32(
                    false, a, false, b, (short)0, acc, false, false);
        }

        // C/D layout: VGPR v, lanes 0-15 -> M=v; lanes 16-31 -> M=v+8; N=l15.
        #pragma unroll
        for (int v = 0; v < 8; ++v) {
            int m   = v + half * 8;
            int row = srow[m];
            if (row >= 0) {
                out[(size_t)row * D_OUT + n0 + l15] = acc[v] + bias;
            }
        }
    }
}

extern "C" void kernel_launch(void* const* d_in, const int* in_sizes, int n_in,
                              void* d_out, int out_size, void* d_ws, size_t ws_size,
                              hipStream_t stream) {
    const float* x      = (const float*)d_in[0];
    const float* onehot = (const float*)d_in[1];
    const float* W      = (const float*)d_in[2];
    const float* Bw     = (const float*)d_in[3];
    float*       out    = (float*)d_out;

    int* counts  = (int*)d_ws;            // 64 ints
    int* rowlist = counts + NEXP;         // 64 * 4096 ints (1 MB)

    dd_zero_counts<<<1, NEXP, 0, stream>>>(counts);
    dd_scatter<<<B_ROWS / 256, 256, 0, stream>>>(onehot, counts, rowlist);
    dim3 grid(NEXP, D_OUT / 64);
    dd_grouped_gemm<<<grid, 128, 0, stream>>>(x, W, Bw, counts, rowlist, out);
}